// ModuleNet_20366734917826
// MI455X (gfx1250) — compile-verified
//
#include <hip/hip_runtime.h>
#include <hip/hip_bf16.h>
#include <stdint.h>

// ---------------- problem constants ----------------
#define BATCH    8192
#define EMBED    128
#define NUM_MOD  64
#define HIDDEN   512
#define OUT_DIM  64
#define MAX_TILES 640                 // <= 64 + 8192/16
#define WCOMB    0.2857142857142857f  // 1/(PATH_LEN*ALPHA) = 2/7

typedef float v2f __attribute__((ext_vector_type(2)));
typedef float v8f __attribute__((ext_vector_type(8)));

// D = A(16x4) * B(4x16) + C, all fp32, wave32 WMMA.
__device__ __forceinline__ v8f wmma_f32_4(v2f a, v2f b, v8f c) {
    return __builtin_amdgcn_wmma_f32_16x16x4_f32(
        /*neg_a=*/false, a, /*neg_b=*/false, b,
        /*c_mod=*/(short)0, c, /*reuse_a=*/false, /*reuse_b=*/false);
}

// ---------------- binning kernels (module ids static) -----------------------
__global__ void zero_counts_kernel(int* counts) {
    if (threadIdx.x < 3 * NUM_MOD) counts[threadIdx.x] = 0;
}

__global__ void count_kernel(const int* __restrict__ mids, int* __restrict__ counts) {
    int b = blockIdx.x * blockDim.x + threadIdx.x;
    if (b < BATCH) {
        for (int h = 0; h < 3; ++h)
            atomicAdd(&counts[h * NUM_MOD + mids[b * 3 + h]], 1);
    }
}

__global__ void scan_build_kernel(const int* __restrict__ counts,
                                  int* __restrict__ offsets, int* __restrict__ cursor,
                                  int4* __restrict__ tiletab, int* __restrict__ ntiles) {
    if (blockIdx.x != 0 || threadIdx.x != 0) return;
    for (int h = 0; h < 3; ++h) {
        int run = 0;
        for (int m = 0; m < NUM_MOD; ++m) {
            offsets[h * NUM_MOD + m] = run;
            cursor [h * NUM_MOD + m] = run;
            run += counts[h * NUM_MOD + m];
        }
        int idx = 0;
        for (int m = 0; m < NUM_MOD; ++m) {
            int c  = counts [h * NUM_MOD + m];
            int st = offsets[h * NUM_MOD + m];
            for (int t = 0; t < c; t += 16) {
                int rem = c - t; if (rem > 16) rem = 16;
                tiletab[h * MAX_TILES + idx] = make_int4(m, st + t, rem, 0);
                ++idx;
            }
        }
        ntiles[h] = idx;
    }
}

__global__ void scatter_kernel(const int* __restrict__ mids,
                               int* __restrict__ cursor, int* __restrict__ rowlist) {
    int b = blockIdx.x * blockDim.x + threadIdx.x;
    if (b < BATCH) {
        for (int h = 0; h < 3; ++h) {
            int m = mids[b * 3 + h];
            int p = atomicAdd(&cursor[h * NUM_MOD + m], 1);
            rowlist[h * BATCH + p] = b;
        }
    }
}

// ---------------- one-time weight packing into WMMA B-fragment order --------
// For out-tile `ot`, k-step k4 (K chunk of 4), lane l (n=l&15, c0=(l>>4)*2):
//   b.x = W[o=ot*16+n][i=4*k4+c0],  b.y = W[o][i+1]
// Two k-steps (k4g = k4>>1, ph = k4&1) are packed per lane into one float4 so the
// GEMM inner loop issues a single coalesced global_load_b128 per two WMMAs.
//   WF[(tile_base + k4g)*128 + l*4 + ph*2 + {0,1}]

__global__ void pack_wm_kernel(const float* __restrict__ W, float* __restrict__ WF) {
    int t = blockIdx.x * blockDim.x + threadIdx.x;   // 64*8*32*32 threads
    if (t >= NUM_MOD * 8 * 32 * 32) return;
    int l  = t & 31;
    int k4 = (t >> 5) & 31;
    int ot = (t >> 10) & 7;
    int m  = t >> 13;
    int n = l & 15, c0 = (l >> 4) * 2;
    int o = ot * 16 + n;
    int i = 4 * k4 + c0;
    const float* src = W + ((size_t)m * EMBED + o) * EMBED + i;
    int k4g = k4 >> 1, ph = k4 & 1;
    float* dst = WF + ((size_t)((m * 8 + ot) * 16 + k4g)) * 128 + l * 4 + ph * 2;
    dst[0] = src[0];
    dst[1] = src[1];
}

__global__ void pack_w1_kernel(const float* __restrict__ W1, float* __restrict__ WF) {
    int t = blockIdx.x * blockDim.x + threadIdx.x;   // 32*32*32 threads
    if (t >= 32 * 32 * 32) return;
    int l  = t & 31;
    int k4 = (t >> 5) & 31;
    int ot = t >> 10;                                 // 0..31
    int n = l & 15, c0 = (l >> 4) * 2;
    int o = ot * 16 + n;
    int i = 4 * k4 + c0;
    const float* src = W1 + (size_t)o * EMBED + i;
    int k4g = k4 >> 1, ph = k4 & 1;
    float* dst = WF + ((size_t)(ot * 16 + k4g)) * 128 + l * 4 + ph * 2;
    dst[0] = src[0];
    dst[1] = src[1];
}

__global__ void pack_w2_kernel(const float* __restrict__ W2, float* __restrict__ WF) {
    int t = blockIdx.x * blockDim.x + threadIdx.x;   // 4*128*32 threads
    if (t >= 4 * 128 * 32) return;
    int l  = t & 31;
    int k4 = (t >> 5) & 127;
    int ot = t >> 12;                                 // 0..3
    int n = l & 15, c0 = (l >> 4) * 2;
    int o = ot * 16 + n;
    int i = 4 * k4 + c0;
    const float* src = W2 + (size_t)o * HIDDEN + i;
    int k4g = k4 >> 1, ph = k4 & 1;
    float* dst = WF + ((size_t)(ot * 64 + k4g)) * 128 + l * 4 + ph * 2;
    dst[0] = src[0];
    dst[1] = src[1];
}

// ---------------- hop GEMM: out[r,o] = relu(sum_i X[r,i]*Wm[o,i] + bias[r,o]) ---
__global__ __launch_bounds__(256) void hop_gemm_kernel(
    const float* __restrict__ embeds, const float* __restrict__ WmF,
    const int*  __restrict__ eids,    const int4* __restrict__ tiletab,
    const int*  __restrict__ ntiles,  const int*  __restrict__ rowlist,
    const float* __restrict__ xin,    float* __restrict__ xout, int hop)
{
    __shared__ float sX[16][132];   // stride%64==4 -> conflict-free ds_load_2addr_b64
    __shared__ float sB[16][132];
    __shared__ int   sRow[16];

    int nt = ntiles[hop];
    if ((int)blockIdx.x >= nt) return;
    int4 tt = tiletab[hop * MAX_TILES + blockIdx.x];
    int m = tt.x, start = tt.y, cnt = tt.z;
    int tid = threadIdx.x;

    if (tid < 16) {
        int slot = (tid < cnt) ? tid : 0;  // pad slots duplicate first row (never stored)
        sRow[tid] = rowlist[hop * BATCH + start + slot];
    }
    __syncthreads();

    for (int q = tid; q < 16 * 32; q += 256) {
        int r = q >> 5, c4 = q & 31;
        int rowg = sRow[r];
        const float* srcx = (hop == 0)
            ? (embeds + (size_t)eids[rowg * 4 + 0] * EMBED)
            : (xin    + (size_t)rowg * EMBED);
        float4 v = ((const float4*)srcx)[c4];
        sX[r][c4 * 4 + 0] = v.x; sX[r][c4 * 4 + 1] = v.y;
        sX[r][c4 * 4 + 2] = v.z; sX[r][c4 * 4 + 3] = v.w;
        const float* srcb = embeds + (size_t)eids[rowg * 4 + hop + 1] * EMBED;
        float4 bv = ((const float4*)srcb)[c4];
        sB[r][c4 * 4 + 0] = bv.x; sB[r][c4 * 4 + 1] = bv.y;
        sB[r][c4 * 4 + 2] = bv.z; sB[r][c4 * 4 + 3] = bv.w;
    }
    __syncthreads();

    int wave = tid >> 5, lane = tid & 31;
    int half = lane >> 4, n = lane & 15, c0 = half * 2;
    int obase = wave * 16;

    const float4* Bb = (const float4*)(WmF + (size_t)(m * 8 + wave) * 16 * 128) + lane;

    v8f acc = {0.f,0.f,0.f,0.f,0.f,0.f,0.f,0.f};
    #pragma unroll
    for (int k4g = 0; k4g < 16; ++k4g) {
        float4 w = Bb[k4g * 32];
        int k = k4g * 8;
        v2f a0, a1, b0, b1f;
        a0.x = sX[n][k + c0];       a0.y = sX[n][k + c0 + 1];
        a1.x = sX[n][k + 4 + c0];   a1.y = sX[n][k + 4 + c0 + 1];
        b0.x = w.x; b0.y = w.y;
        b1f.x = w.z; b1f.y = w.w;
        acc = wmma_f32_4(a0, b0,  acc);
        acc = wmma_f32_4(a1, b1f, acc);
    }

    for (int r = 0; r < 8; ++r) {
        int row = r + half * 8;            // D layout: VGPR r -> M=r (lanes<16) / r+8
        if (row < cnt) {
            int rowg = sRow[row];
            int o = obase + n;
            float bias = sB[row][o];
            float v = acc[r] + bias;
            v = v > 0.f ? v : 0.f;
            float res = (hop == 2) ? ((1.f - WCOMB) * bias + WCOMB * v) : v;
            xout[(size_t)rowg * EMBED + o] = res;
        }
    }
}

// ---------------- fused MLP: logits = relu(X@W1^T+b1)@W2^T + b2 -----------------
__global__ __launch_bounds__(256) void mlp_kernel(
    const float* __restrict__ xin, const float* __restrict__ W1F, const float* __restrict__ b1,
    const float* __restrict__ W2F, const float* __restrict__ b2, float* __restrict__ out)
{
    __shared__ float sX[16][132];
    __shared__ float sH[16][516];   // 516%64==4 -> conflict-free

    int bt = blockIdx.x;            // 16-row tile of the batch
    int tid = threadIdx.x;
    int wave = tid >> 5, lane = tid & 31;
    int half = lane >> 4, n = lane & 15, c0 = half * 2;

    for (int q = tid; q < 16 * 32; q += 256) {
        int r = q >> 5, c4 = q & 31;
        float4 v = ((const float4*)(xin + (size_t)(bt * 16 + r) * EMBED))[c4];
        sX[r][c4 * 4 + 0] = v.x; sX[r][c4 * 4 + 1] = v.y;
        sX[r][c4 * 4 + 2] = v.z; sX[r][c4 * 4 + 3] = v.w;
    }
    __syncthreads();

    // phase 1: 512 hidden outs = 4 passes x 8 waves x 16-wide tiles, K=128
    for (int p = 0; p < 4; ++p) {
        int ot = p * 8 + wave;                      // 0..31
        int ob = ot * 16;
        const float4* Bb = (const float4*)(W1F + (size_t)ot * 16 * 128) + lane;
        v8f acc = {0.f,0.f,0.f,0.f,0.f,0.f,0.f,0.f};
        #pragma unroll
        for (int k4g = 0; k4g < 16; ++k4g) {
            float4 w = Bb[k4g * 32];
            int k = k4g * 8;
            v2f a0, a1, b0, b1f;
            a0.x = sX[n][k + c0];       a0.y = sX[n][k + c0 + 1];
            a1.x = sX[n][k + 4 + c0];   a1.y = sX[n][k + 4 + c0 + 1];
            b0.x = w.x; b0.y = w.y;
            b1f.x = w.z; b1f.y = w.w;
            acc = wmma_f32_4(a0, b0,  acc);
            acc = wmma_f32_4(a1, b1f, acc);
        }
        float bj = b1[ob + n];
        for (int r = 0; r < 8; ++r) {
            int row = r + half * 8;
            float v = acc[r] + bj;
            sH[row][ob + n] = v > 0.f ? v : 0.f;
        }
    }
    __syncthreads();

    // phase 2: 64 logits = 4 waves x 16-wide tiles, K=512
    if (wave < 4) {
        int ob = wave * 16;
        const float4* Bb = (const float4*)(W2F + (size_t)wave * 64 * 128) + lane;
        v8f acc = {0.f,0.f,0.f,0.f,0.f,0.f,0.f,0.f};
        #pragma unroll 8
        for (int k4g = 0; k4g < 64; ++k4g) {
            float4 w = Bb[k4g * 32];
            int k = k4g * 8;
            v2f a0, a1, b0, b1f;
            a0.x = sH[n][k + c0];       a0.y = sH[n][k + c0 + 1];
            a1.x = sH[n][k + 4 + c0];   a1.y = sH[n][k + 4 + c0 + 1];
            b0.x = w.x; b0.y = w.y;
            b1f.x = w.z; b1f.y = w.w;
            acc = wmma_f32_4(a0, b0,  acc);
            acc = wmma_f32_4(a1, b1f, acc);
        }
        float bo = b2[ob + n];
        for (int r = 0; r < 8; ++r) {
            int row = r + half * 8;
            out[(size_t)(bt * 16 + row) * OUT_DIM + ob + n] = acc[r] + bo;
        }
    }
}

// ---------------- host launcher ----------------
extern "C" void kernel_launch(void* const* d_in, const int* in_sizes, int n_in,
                              void* d_out, int out_size, void* d_ws, size_t ws_size,
                              hipStream_t stream) {
    const float* embeds = (const float*)d_in[0];
    const float* Wmods  = (const float*)d_in[1];
    const float* W1     = (const float*)d_in[2];
    const float* b1     = (const float*)d_in[3];
    const float* W2     = (const float*)d_in[4];
    const float* b2     = (const float*)d_in[5];
    const int*   eids   = (const int*)d_in[6];
    const int*   mids   = (const int*)d_in[7];
    float*       out    = (float*)d_out;

    char* ws = (char*)d_ws;
    size_t off = 0;
    float* X0      = (float*)(ws + off); off += (size_t)BATCH * EMBED * 4;
    float* X1      = (float*)(ws + off); off += (size_t)BATCH * EMBED * 4;
    float* WmF     = (float*)(ws + off); off += (size_t)NUM_MOD * EMBED * EMBED * 4;
    float* W1F     = (float*)(ws + off); off += (size_t)HIDDEN * EMBED * 4;
    float* W2F     = (float*)(ws + off); off += (size_t)OUT_DIM * HIDDEN * 4;
    int*   counts  = (int*)(ws + off);   off += 3 * NUM_MOD * 4;
    int*   offsets = (int*)(ws + off);   off += 3 * NUM_MOD * 4;
    int*   cursor  = (int*)(ws + off);   off += 3 * NUM_MOD * 4;
    int*   rowlist = (int*)(ws + off);   off += 3 * BATCH * 4;
    int4*  tiletab = (int4*)(ws + off);  off += 3 * MAX_TILES * sizeof(int4);
    int*   ntiles  = (int*)(ws + off);   off += 64;
    (void)ws_size; (void)in_sizes; (void)n_in; (void)out_size;

    // binning
    zero_counts_kernel<<<1, 256, 0, stream>>>(counts);
    count_kernel<<<BATCH / 256, 256, 0, stream>>>(mids, counts);
    scan_build_kernel<<<1, 32, 0, stream>>>(counts, offsets, cursor, tiletab, ntiles);
    scatter_kernel<<<BATCH / 256, 256, 0, stream>>>(mids, cursor, rowlist);

    // one-time weight packing into WMMA fragment order (L2-resident afterwards)
    pack_wm_kernel<<<(NUM_MOD * 8 * 32 * 32) / 256, 256, 0, stream>>>(Wmods, WmF);
    pack_w1_kernel<<<(32 * 32 * 32) / 256, 256, 0, stream>>>(W1, W1F);
    pack_w2_kernel<<<(4 * 128 * 32) / 256, 256, 0, stream>>>(W2, W2F);

    // three modulated hops as per-module WMMA GEMMs
    hop_gemm_kernel<<<MAX_TILES, 256, 0, stream>>>(embeds, WmF, eids, tiletab, ntiles, rowlist, X1, X0, 0);
    hop_gemm_kernel<<<MAX_TILES, 256, 0, stream>>>(embeds, WmF, eids, tiletab, ntiles, rowlist, X0, X1, 1);
    hop_gemm_kernel<<<MAX_TILES, 256, 0, stream>>>(embeds, WmF, eids, tiletab, ntiles, rowlist, X1, X0, 2);

    // fused two-layer MLP
    mlp_kernel<<<BATCH / 16, 256, 0, stream>>>(X0, W1F, b1, W2F, b2, out);
}